// QuanvolutionDualClassifier_65481071406131
// MI455X (gfx1250) — compile-verified
//
#include <hip/hip_runtime.h>
#include <hip/hip_bf16.h>
#include <math.h>

// CDNA5 / gfx1250, wave32. Full-fp32 WMMA path via V_WMMA_F32_16X16X4_F32.

typedef __attribute__((ext_vector_type(2))) float v2f;
typedef __attribute__((ext_vector_type(4))) float v4f;
typedef __attribute__((ext_vector_type(8))) float v8f;

#define B_SZ      4096
#define FEAT      1568
#define NPATCH    196                 // 14*14 patches per image
#define TOT_PATCH (B_SZ * NPATCH)     // 802816 == 16 * 50176
#define NTILE     (TOT_PATCH / 16)    // 50176, divisible by 8

// Exchange value with the other 16-lane half: ds_swizzle SWAPX16
// (group-of-32 mode: xor_mask=0x10 -> offset = (0x10<<10) | 0x1f).
__device__ __forceinline__ float swap_half(float v) {
  return __int_as_float(
      __builtin_amdgcn_ds_swizzle(__float_as_int(v), 0x401f));
}

// ---------------------------------------------------------------------------
// Classical branch: depthwise 3x3 s2 p1 conv + pointwise scale, *0.5 (attn
// block is identically sigmoid(0)=0.5).  feats[b, c*196 + pp] for c=0..3.
// ---------------------------------------------------------------------------
__global__ void classical_kernel(const float* __restrict__ x,
                                 const float* __restrict__ dw,
                                 const float* __restrict__ pw,
                                 float* __restrict__ feats) {
  int idx = blockIdx.x * blockDim.x + threadIdx.x;
  if (idx >= TOT_PATCH) return;
  int b = idx / NPATCH, pp = idx % NPATCH;
  int pr = pp / 14, pc = pp % 14;
  const float* xb = x + (size_t)b * 784;
  float acc = 0.f;
#pragma unroll
  for (int kr = 0; kr < 3; ++kr) {
    int r = 2 * pr - 1 + kr;
    if ((unsigned)r >= 28u) continue;
#pragma unroll
    for (int kc = 0; kc < 3; ++kc) {
      int c = 2 * pc - 1 + kc;
      if ((unsigned)c >= 28u) continue;
      acc = fmaf(xb[r * 28 + c], dw[kr * 3 + kc], acc);
    }
  }
  float* fr = feats + (size_t)b * FEAT;
#pragma unroll
  for (int c = 0; c < 4; ++c) fr[c * NPATCH + pp] = 0.5f * pw[c] * acc;
}

// ---------------------------------------------------------------------------
// Quantum branch, one wave per 16-patch tile, computed TRANSPOSED:
//   psi^T = U @ amp^T    (A = U rows, B = amp columns built from sincos)
// D layout: rows = output states (VGPR v -> state v + 8*hi), columns =
// patches (lane n). Pauli-Z expectations are then per-lane sums over the 8
// accumulator regs with compile-time signs (state bits 2..0 == bits of v,
// state bit 3 == lane half), plus one XOR-16 half-swap per wire, and each
// low-half lane stores its patch's 4 ev values as one b128 store.
// ---------------------------------------------------------------------------
__global__ void __launch_bounds__(256)
quantum_kernel(const float* __restrict__ x,
               const float* __restrict__ u_re,
               const float* __restrict__ u_im,
               float* __restrict__ feats) {
  int lane = threadIdx.x & 31;
  int wv   = threadIdx.x >> 5;
  int tile = blockIdx.x * 8 + wv;         // 0 .. NTILE-1 exactly
  int hi   = lane >> 4;                   // which 16-lane half
  int n    = lane & 15;                   // A row (M) / B,C,D column (N)

  // ---- angles for the patch in this lane's B column (and D column) ----
  int P  = tile * 16 + n;
  int b  = P / NPATCH, pp = P % NPATCH;
  int pr = pp / 14,    pc = pp % 14;
  const float* xp = x + (size_t)b * 784 + (2 * pr) * 28 + 2 * pc;
  v2f x01 = *(const v2f*)(xp);            // 8B aligned: all offsets even
  v2f x23 = *(const v2f*)(xp + 28);
  float s0, c0, s1, c1, s2, c2, s3, c3;
  __sincosf(0.5f * x01.x, &s0, &c0);      // wire 0 -> state bit 3 (MSB)
  __sincosf(0.5f * x01.y, &s1, &c1);      // wire 1 -> bit 2
  __sincosf(0.5f * x23.x, &s2, &c2);      // wire 2 -> bit 1
  __sincosf(0.5f * x23.y, &s3, &c3);      // wire 3 -> bit 0

  auto ampv = [&](int s) -> float {
    return ((s & 8) ? s0 : c0) * ((s & 4) ? s1 : c1) *
           ((s & 2) ? s2 : c2) * ((s & 1) ? s3 : c3);
  };

  // ---- psi^T = U @ amp^T via 4 chained K=4 f32 WMMAs (re and im) ----
  v8f pre = {}; v8f pim = {};
#pragma unroll
  for (int kk = 0; kk < 4; ++kk) {
    int k0 = kk * 4 + (hi ? 2 : 0);       // A/B layout: upper half holds K+2
    // A[M=n, k] = U[n, k] = u[n*16 + k]; (k0, k0+1) consecutive -> v2f load
    v2f ar = *(const v2f*)(u_re + n * 16 + k0);
    v2f ai = *(const v2f*)(u_im + n * 16 + k0);
    // B[k, N=n] = amp^T[k, patch n] = amp(patch n, state k)
    v2f ba; ba.x = ampv(k0); ba.y = ampv(k0 + 1);
    pre = __builtin_amdgcn_wmma_f32_16x16x4_f32(false, ar, false, ba,
                                                (short)0, pre, false, false);
    pim = __builtin_amdgcn_wmma_f32_16x16x4_f32(false, ai, false, ba,
                                                (short)0, pim, false, false);
  }

  // ---- probs -> Z expectations: in-lane signed sums + one half-swap ----
  float st = 0.f, w1 = 0.f, w2 = 0.f, w3 = 0.f;
#pragma unroll
  for (int v = 0; v < 8; ++v) {
    float p = pre[v] * pre[v] + pim[v] * pim[v];   // probs[state v+8*hi, n]
    st += p;                                       // wire0 sign handled below
    w1 += (v & 4) ? -p : p;                        // state bit2
    w2 += (v & 2) ? -p : p;                        // state bit1
    w3 += (v & 1) ? -p : p;                        // state bit0
  }
  float e0 = hi ? -st : st;                        // state bit3 == lane half
  e0 += swap_half(e0);
  float e1 = w1 + swap_half(w1);
  float e2 = w2 + swap_half(w2);
  float e3 = w3 + swap_half(w3);

  if (hi == 0) {                                   // one b128 store per patch
    v4f ev = {e0, e1, e2, e3};
    *(v4f*)(feats + (size_t)b * FEAT + 784 + pp * 4) = ev;  // 16B aligned
  }
}

// ---------------------------------------------------------------------------
// Head GEMM: logits = feats @ lin_w^T + b.  M=4096, N=10 (padded to 16),
// K=1568 as 392 chained f32 WMMAs per 16-row tile.
// ---------------------------------------------------------------------------
__global__ void __launch_bounds__(256)
linear_kernel(const float* __restrict__ feats,
              const float* __restrict__ lw,
              const float* __restrict__ lb,
              float* __restrict__ logits) {
  int lane = threadIdx.x & 31;
  int wv   = threadIdx.x >> 5;
  int tile = blockIdx.x * 8 + wv;         // 0..255
  int hi   = lane >> 4;
  int n    = lane & 15;

  const float* fa  = feats + (size_t)(tile * 16 + n) * FEAT;  // A row M = n
  int   nb  = (n < 10) ? n : 9;                                // clamp (no OOB)
  float msk = (n < 10) ? 1.f : 0.f;                            // pad N 10->16
  const float* lwr = lw + (size_t)nb * FEAT;

  v8f acc = {};
  for (int k = 0; k < FEAT; k += 4) {
    int k0 = k + (hi ? 2 : 0);
    v2f a  = *(const v2f*)(fa + k0);
    v2f bw = *(const v2f*)(lwr + k0);
    bw.x *= msk; bw.y *= msk;
    acc = __builtin_amdgcn_wmma_f32_16x16x4_f32(false, a, false, bw,
                                                (short)0, acc, false, false);
  }
#pragma unroll
  for (int v = 0; v < 8; ++v) {
    if (n < 10) {
      int row = tile * 16 + v + (hi ? 8 : 0);
      logits[row * 10 + n] = acc[v] + lb[n];
    }
  }
}

// ---------------------------------------------------------------------------
// log_softmax over 10 logits per row.
// ---------------------------------------------------------------------------
__global__ void lsm_kernel(const float* __restrict__ logits,
                           float* __restrict__ out) {
  int r = blockIdx.x * blockDim.x + threadIdx.x;
  if (r >= B_SZ) return;
  const float* l = logits + r * 10;
  float m = l[0];
#pragma unroll
  for (int j = 1; j < 10; ++j) m = fmaxf(m, l[j]);
  float s = 0.f;
#pragma unroll
  for (int j = 0; j < 10; ++j) s += __expf(l[j] - m);
  float ls = __logf(s);
  float* o = out + r * 10;
#pragma unroll
  for (int j = 0; j < 10; ++j) o[j] = l[j] - m - ls;
}

// ---------------------------------------------------------------------------
extern "C" void kernel_launch(void* const* d_in, const int* in_sizes, int n_in,
                              void* d_out, int out_size, void* d_ws, size_t ws_size,
                              hipStream_t stream) {
  const float* x   = (const float*)d_in[0];   // (4096,1,28,28)
  const float* dw  = (const float*)d_in[1];   // (1,1,3,3)
  const float* pw  = (const float*)d_in[2];   // (4,1,1,1)
  // d_in[3], d_in[4]: zero-size attention weights -> branch is constant 0.5
  const float* ure = (const float*)d_in[5];   // (16,16)
  const float* uim = (const float*)d_in[6];   // (16,16)
  const float* lw  = (const float*)d_in[7];   // (10,1568)
  const float* lb  = (const float*)d_in[8];   // (10,)

  // workspace: feats (4096x1568 f32) + logits (4096x10 f32) ~= 25.9 MB
  float* feats  = (float*)d_ws;
  float* logits = feats + (size_t)B_SZ * FEAT;

  classical_kernel<<<TOT_PATCH / 256, 256, 0, stream>>>(x, dw, pw, feats);
  quantum_kernel<<<NTILE / 8, 256, 0, stream>>>(x, ure, uim, feats);
  linear_kernel<<<256 / 8, 256, 0, stream>>>(feats, lw, lb, logits);
  lsm_kernel<<<(B_SZ + 255) / 256, 256, 0, stream>>>(logits, (float*)d_out);
}